// ImgCap_12498354831512
// MI455X (gfx1250) — compile-verified
//
#include <hip/hip_runtime.h>
#include <math.h>
#include <stdint.h>

// ---------------- CDNA5 WMMA types (fp32 16x16x4) ----------------
typedef float v2f __attribute__((ext_vector_type(2)));
typedef float v8f __attribute__((ext_vector_type(8)));

// ---------------- async global->LDS (gfx1250), guarded ----------------
#if defined(__gfx1250__) && \
    __has_builtin(__builtin_amdgcn_global_load_async_to_lds_b32) && \
    __has_builtin(__builtin_amdgcn_global_load_async_to_lds_b128) && \
    __has_builtin(__builtin_amdgcn_s_wait_asynccnt)
#define USE_ASYNC_LDS 1
#else
#define USE_ASYNC_LDS 0
#endif

#if USE_ASYNC_LDS
typedef int vi4 __attribute__((vector_size(4 * sizeof(int))));
typedef __attribute__((address_space(1))) int gi32_t;
typedef __attribute__((address_space(3))) int li32_t;
typedef __attribute__((address_space(1))) vi4 gvi4_t;
typedef __attribute__((address_space(3))) vi4 lvi4_t;
__device__ __forceinline__ void async_g2l_b32(const float* g, float* l) {
    __builtin_amdgcn_global_load_async_to_lds_b32(
        (gi32_t*)(uintptr_t)g, (li32_t*)l, 0, 0);
}
__device__ __forceinline__ void async_g2l_b128(const float* g, float* l) {
    __builtin_amdgcn_global_load_async_to_lds_b128(
        (gvi4_t*)(uintptr_t)g, (lvi4_t*)l, 0, 0);
}
__device__ __forceinline__ void async_wait0() {
    __builtin_amdgcn_s_wait_asynccnt(0);
}
#else
__device__ __forceinline__ void async_g2l_b32(const float* g, float* l) { *l = *g; }
__device__ __forceinline__ void async_g2l_b128(const float* g, float* l) {
    *(float4*)l = *(const float4*)g;
}
__device__ __forceinline__ void async_wait0() {}
#endif

// ---------------- workspace layout (float offsets) ----------------
#define WS_POOL1 0                                  // [16][4][256][256]
#define WS_XF    (WS_POOL1 + 16*4*256*256)          // [16][16][16384]
#define WS_ASUM  (WS_XF    + 16*16*16384)           // [16][64]
#define WS_VLADA (WS_ASUM  + 16*64)                 // [16][64][16]
#define WS_VNORM (WS_VLADA + 16*64*16)              // [16][1024]
#define WS_FEAT  (WS_VNORM + 16*1024)               // [16][512]

__device__ __forceinline__ float sigmoidf_(float x) {
    return 1.0f / (1.0f + __expf(-x));
}

// ================= kernel 0: zero atomic accumulators =================
__global__ void k_zero(float* ws) {
    int i = blockIdx.x * blockDim.x + threadIdx.x;
    int n = 16*64 + 16*64*16;                 // a_sum + vlad_acc
    if (i < n) ws[WS_ASUM + i] = 0.0f;
}

// ================= kernel 1: conv1(3->4,3x3,SAME)+ReLU+avgpool2 =================
// img [16][3][512][512] -> pool1 [16][4][256][256]
// block = pooled tile 64x4, all 4 ocs per thread; input tile staged in LDS once.
#define IS1 132
__global__ void k_conv1(const float* __restrict__ img,
                        const float* __restrict__ w,   // [4][3][3][3]
                        const float* __restrict__ b,   // [4]
                        float* __restrict__ out) {
    __shared__ float T[3 * 10 * IS1];   // 3960 floats
    __shared__ float Wl[108];
    __shared__ float Bl[4];
    const int tid = threadIdx.x;
    if (tid < 108) Wl[tid] = w[tid];
    if (tid < 4)   Bl[tid] = b[tid];

    const int bx = blockIdx.x;          // 0..3
    const int by = blockIdx.y;          // 0..63
    const int n  = blockIdx.z;          // 0..15
    const int x0 = bx * 128 - 1;
    const int y0 = by * 8 - 1;

    const float* imgn = img + (size_t)n * 3 * 512 * 512;
    for (int j = tid; j < 3 * 10 * 130; j += 256) {
        int c  = j % 130;
        int r  = (j / 130) % 10;
        int ic = j / 1300;
        int y = y0 + r, x = x0 + c;
        float* dst = &T[(ic * 10 + r) * IS1 + c];
        if (y >= 0 && y < 512 && x >= 0 && x < 512)
            async_g2l_b32(&imgn[(ic * 512 + y) * 512 + x], dst);
        else
            *dst = 0.0f;
    }
    async_wait0();
    __syncthreads();

    const int pxl = tid & 63, pyl = tid >> 6;
    const int px = bx * 64 + pxl, py = by * 4 + pyl;

    float p[3][4][4];
#pragma unroll
    for (int ic = 0; ic < 3; ++ic)
#pragma unroll
        for (int r = 0; r < 4; ++r)
#pragma unroll
            for (int c = 0; c < 4; ++c)
                p[ic][r][c] = T[(ic * 10 + 2 * pyl + r) * IS1 + 2 * pxl + c];

#pragma unroll
    for (int oc = 0; oc < 4; ++oc) {
        float wr[27];
#pragma unroll
        for (int i = 0; i < 27; ++i) wr[i] = Wl[oc * 27 + i];
        float s = 0.0f;
#pragma unroll
        for (int dy = 0; dy < 2; ++dy)
#pragma unroll
            for (int dx = 0; dx < 2; ++dx) {
                float acc = Bl[oc];
#pragma unroll
                for (int ic = 0; ic < 3; ++ic)
#pragma unroll
                    for (int ky = 0; ky < 3; ++ky)
#pragma unroll
                        for (int kx = 0; kx < 3; ++kx)
                            acc += p[ic][dy + ky][dx + kx] * wr[(ic * 3 + ky) * 3 + kx];
                s += fmaxf(acc, 0.0f);
            }
        out[((n * 4 + oc) * 256 + py) * 256 + px] = s * 0.25f;
    }
}

// ================= kernel 2: conv2(4->16,3x3,SAME)+ReLU+avgpool2 =================
// pool1 [16][4][256][256] -> xf [16][16][16384]  (h = y*128+x)
#define IS2 132
__global__ void k_conv2(const float* __restrict__ in,
                        const float* __restrict__ w,   // [16][4][3][3]
                        const float* __restrict__ b,   // [16]
                        float* __restrict__ xf) {
    __shared__ float T[4 * 10 * IS2];   // 5280 floats
    __shared__ float Wl[576];
    __shared__ float Bl[16];
    const int tid = threadIdx.x;
    for (int j = tid; j < 576; j += 256) Wl[j] = w[j];
    if (tid < 16) Bl[tid] = b[tid];

    const int bx = blockIdx.x;          // 0..1
    const int by = blockIdx.y;          // 0..31
    const int n  = blockIdx.z;          // 0..15
    const int x0 = bx * 128 - 1;
    const int y0 = by * 8 - 1;

    const float* inn = in + (size_t)n * 4 * 256 * 256;
    for (int j = tid; j < 4 * 10 * 130; j += 256) {
        int c  = j % 130;
        int r  = (j / 130) % 10;
        int ic = j / 1300;
        int y = y0 + r, x = x0 + c;
        float* dst = &T[(ic * 10 + r) * IS2 + c];
        if (y >= 0 && y < 256 && x >= 0 && x < 256)
            async_g2l_b32(&inn[(ic * 256 + y) * 256 + x], dst);
        else
            *dst = 0.0f;
    }
    async_wait0();
    __syncthreads();

    const int pxl = tid & 63, pyl = tid >> 6;
    const int px = bx * 64 + pxl, py = by * 4 + pyl;

    float p[4][4][4];
#pragma unroll
    for (int ic = 0; ic < 4; ++ic)
#pragma unroll
        for (int r = 0; r < 4; ++r)
#pragma unroll
            for (int c = 0; c < 4; ++c)
                p[ic][r][c] = T[(ic * 10 + 2 * pyl + r) * IS2 + 2 * pxl + c];

    for (int oc = 0; oc < 16; ++oc) {
        float wr[36];
#pragma unroll
        for (int i = 0; i < 36; ++i) wr[i] = Wl[oc * 36 + i];
        float s = 0.0f;
#pragma unroll
        for (int dy = 0; dy < 2; ++dy)
#pragma unroll
            for (int dx = 0; dx < 2; ++dx) {
                float acc = Bl[oc];
#pragma unroll
                for (int ic = 0; ic < 4; ++ic)
#pragma unroll
                    for (int ky = 0; ky < 3; ++ky)
#pragma unroll
                        for (int kx = 0; kx < 3; ++kx)
                            acc += p[ic][dy + ky][dx + kx] * wr[(ic * 3 + ky) * 3 + kx];
                s += fmaxf(acc, 0.0f);
            }
        xf[((n * 16 + oc) << 14) + py * 128 + px] = s * 0.25f;
    }
}

// ================= kernel 3: fused NetVLAD chunk (WMMA fp32) =================
// grid (128 chunks of 128 px, 16 images), 256 threads (8 wave32)
#define XS 136   // Xs row stride  (136 % 64 == 8 -> disjoint half-wave banks)
#define SS 131   // Ss row stride  (odd)
#define WSD 20   // Ws row stride  (16B aligned rows; 20m mod 64 conflict-free)
__global__ void k_netvlad(const float* __restrict__ xf,
                          const float* __restrict__ nv_w,   // [64][16]
                          const float* __restrict__ nv_b,   // [64]
                          float* __restrict__ a_sum,        // [16][64]
                          float* __restrict__ vlad_acc) {   // [16][64][16]
    __shared__ float Xs[16 * XS];
    __shared__ float Ssm[64 * SS];
    __shared__ float Ws[64 * WSD];
    __shared__ float Bsh[64];

    const int tid  = threadIdx.x;
    const int lane = tid & 31;
    const int wv   = tid >> 5;
    const int mrow = lane & 15;
    const int half = lane >> 4;
    const int n    = blockIdx.y;
    const int h0   = blockIdx.x * 128;

    // --- async stage X chunk [16ch x 128px] + weights into LDS ---
#pragma unroll
    for (int i = 0; i < 2; ++i) {
        int v = tid + i * 256;              // 512 vec4 = 2048 floats
        int c = v >> 5, colv = v & 31;
        async_g2l_b128(&xf[((n * 16 + c) << 14) + h0 + colv * 4],
                       &Xs[c * XS + colv * 4]);
    }
    {                                        // 256 vec4 = 1024 floats
        int k = tid >> 2, c4 = (tid & 3) * 4;
        async_g2l_b128(&nv_w[k * 16 + c4], &Ws[k * WSD + c4]);
    }
    if (tid < 64) Bsh[tid] = nv_b[tid];
    async_wait0();
    __syncthreads();

    // --- stage 1: S = W(64x16) @ X(16x128); 32 tiles, 4 per wave ---
    for (int t = wv * 4; t < wv * 4 + 4; ++t) {
        int mt = t & 3, nt = t >> 2;
        int col = nt * 16 + mrow;
        v8f acc = {};
#pragma unroll
        for (int kk = 0; kk < 4; ++kk) {
            int k0 = kk * 4 + 2 * half;
            v2f a, bb;
            a.x  = Ws[(mt * 16 + mrow) * WSD + k0];
            a.y  = Ws[(mt * 16 + mrow) * WSD + k0 + 1];
            bb.x = Xs[k0 * XS + col];
            bb.y = Xs[(k0 + 1) * XS + col];
            acc = __builtin_amdgcn_wmma_f32_16x16x4_f32(
                false, a, false, bb, (short)0, acc, false, false);
        }
#pragma unroll
        for (int v = 0; v < 8; ++v) {
            int kg = mt * 16 + v + 8 * half;
            Ssm[kg * SS + col] = acc[v] + Bsh[kg];
        }
    }
    __syncthreads();

    // --- stage 2: softmax over 64 clusters, one column per thread ---
    if (tid < 128) {
        float mx = -3.4e38f;
#pragma unroll 8
        for (int k = 0; k < 64; ++k) mx = fmaxf(mx, Ssm[k * SS + tid]);
        float sum = 0.0f;
#pragma unroll 8
        for (int k = 0; k < 64; ++k) {
            float e = __expf(Ssm[k * SS + tid] - mx);
            Ssm[k * SS + tid] = e;
            sum += e;
        }
        float inv = 1.0f / sum;
#pragma unroll 8
        for (int k = 0; k < 64; ++k) Ssm[k * SS + tid] *= inv;
    }
    __syncthreads();

    // --- stage 3a: a_sum partials ---
    if (tid < 128) {
        int k = tid & 63, part = tid >> 6;
        float s = 0.0f;
#pragma unroll 8
        for (int c = 0; c < 64; ++c) s += Ssm[k * SS + part * 64 + c];
        atomicAdd(&a_sum[n * 64 + k], s);
    }

    // --- stage 3b: V_partial = S(64x128) @ X^T(128x16) ---
    {
        int mt = wv & 3;
        int kh = (wv >> 2) * 64;
        v8f acc = {};
        for (int ks = 0; ks < 16; ++ks) {
            int hh = kh + ks * 4 + 2 * half;
            v2f a, bb;
            a.x  = Ssm[(mt * 16 + mrow) * SS + hh];
            a.y  = Ssm[(mt * 16 + mrow) * SS + hh + 1];
            bb.x = Xs[mrow * XS + hh];
            bb.y = Xs[mrow * XS + hh + 1];
            acc = __builtin_amdgcn_wmma_f32_16x16x4_f32(
                false, a, false, bb, (short)0, acc, false, false);
        }
#pragma unroll
        for (int v = 0; v < 8; ++v) {
            int k = mt * 16 + v + 8 * half;
            atomicAdd(&vlad_acc[(n * 64 + k) * 16 + mrow], acc[v]);
        }
    }
}

// ================= kernel 4a: subtract centroids, double L2 norm =================
__global__ void k_vladnorm(const float* __restrict__ vlad_acc,
                           const float* __restrict__ a_sum,
                           const float* __restrict__ centroids, // [64][16]
                           float* __restrict__ vnorm) {          // [16][1024]
    __shared__ float buf[1024];
    __shared__ float red[256];
    int n = blockIdx.x, t = threadIdx.x;

    if (t < 64) {
        float as = a_sum[n * 64 + t];
        float v[16], ss = 0.0f;
#pragma unroll
        for (int c = 0; c < 16; ++c) {
            v[c] = vlad_acc[(n * 64 + t) * 16 + c] - as * centroids[t * 16 + c];
            ss += v[c] * v[c];
        }
        float inv = 1.0f / fmaxf(sqrtf(ss), 1e-12f);
#pragma unroll
        for (int c = 0; c < 16; ++c) buf[t * 16 + c] = v[c] * inv;
    }
    __syncthreads();

    float sq = 0.0f;
#pragma unroll
    for (int i = 0; i < 4; ++i) { float x = buf[t * 4 + i]; sq += x * x; }
    red[t] = sq;
    __syncthreads();
    for (int s = 128; s > 0; s >>= 1) {
        if (t < s) red[t] += red[t + s];
        __syncthreads();
    }
    float inv2 = 1.0f / fmaxf(sqrtf(red[0]), 1e-12f);
#pragma unroll
    for (int i = 0; i < 4; ++i) vnorm[n * 1024 + t * 4 + i] = buf[t * 4 + i] * inv2;
}

// ================= kernel 4b: features = vnorm(16x1024) @ fc_w^T + b (WMMA) ====
__global__ void k_fc(const float* __restrict__ vnorm,   // [16][1024]
                     const float* __restrict__ fc_w,    // [512][1024]
                     const float* __restrict__ fc_b,    // [512]
                     float* __restrict__ feat) {        // [16][512]
    const int lane = threadIdx.x & 31;
    const int wv   = threadIdx.x >> 5;
    const int mrow = lane & 15;
    const int half = lane >> 4;

    for (int nt = wv * 4; nt < wv * 4 + 4; ++nt) {       // 32 col tiles / 8 waves
        int hd = nt * 16 + mrow;
        v8f acc = {};
        for (int ks = 0; ks < 256; ++ks) {
            int kj = ks * 4 + 2 * half;
            v2f a, bb;
            a.x  = vnorm[mrow * 1024 + kj];
            a.y  = vnorm[mrow * 1024 + kj + 1];
            bb.x = fc_w[hd * 1024 + kj];
            bb.y = fc_w[hd * 1024 + kj + 1];
            acc = __builtin_amdgcn_wmma_f32_16x16x4_f32(
                false, a, false, bb, (short)0, acc, false, false);
        }
#pragma unroll
        for (int v = 0; v < 8; ++v) {
            int nrow = v + 8 * half;                      // image index
            feat[nrow * 512 + hd] = acc[v] + fc_b[hd];
        }
    }
}

// ================= kernel 5: sequential RNN decoder + MSE loss =================
__global__ void k_rnn(const float* __restrict__ feat,   // [16][512]; row 0 used
                      const float* __restrict__ cap,    // [16][64][300]; sample 0
                      const float* __restrict__ hid_w,  // [512][812]
                      const float* __restrict__ hid_b,  // [512]
                      const float* __restrict__ out_w,  // [300][512]
                      const float* __restrict__ out_b,  // [300]
                      float* __restrict__ loss_out) {
    __shared__ float hcur[512], hnew[512], xin[304], red[512];
    const int i = threadIdx.x;

    hcur[i] = feat[i];
    if (i < 300) xin[i] = cap[i];
    float lacc = 0.0f;
    __syncthreads();

    for (int t = 1; t < 64; ++t) {
        const float* wr = hid_w + (size_t)i * 812;
        float d = hid_b[i];
        for (int j = 0; j < 300; ++j) d += wr[j] * xin[j];
        for (int j = 0; j < 512; ++j) d += wr[300 + j] * hcur[j];
        hnew[i] = sigmoidf_(d);
        __syncthreads();

        if (i < 300) {
            const float* orr = out_w + (size_t)i * 512;
            float y = out_b[i];
            for (int j = 0; j < 512; ++j) y += orr[j] * hnew[j];
            y = sigmoidf_(y);
            float e = y - cap[t * 300 + i];
            lacc += e * e;
        }
        __syncthreads();

        hcur[i] = hnew[i];
        if (i < 300) xin[i] = cap[t * 300 + i];
        __syncthreads();
    }

    red[i] = lacc;
    __syncthreads();
    for (int s = 256; s > 0; s >>= 1) {
        if (i < s) red[i] += red[i + s];
        __syncthreads();
    }
    if (i == 0) loss_out[0] = red[0] / (float)(64 * 300);
}

// ================= launcher =================
extern "C" void kernel_launch(void* const* d_in, const int* in_sizes, int n_in,
                              void* d_out, int out_size, void* d_ws, size_t ws_size,
                              hipStream_t stream) {
    (void)in_sizes; (void)n_in; (void)out_size; (void)ws_size;
    const float* img       = (const float*)d_in[0];
    const float* cap       = (const float*)d_in[1];
    const float* conv1_w   = (const float*)d_in[2];
    const float* conv1_b   = (const float*)d_in[3];
    const float* conv2_w   = (const float*)d_in[4];
    const float* conv2_b   = (const float*)d_in[5];
    const float* centroids = (const float*)d_in[6];
    const float* nv_w      = (const float*)d_in[7];
    const float* nv_b      = (const float*)d_in[8];
    const float* fc_w      = (const float*)d_in[9];
    const float* fc_b      = (const float*)d_in[10];
    const float* hid_w     = (const float*)d_in[11];
    const float* hid_b     = (const float*)d_in[12];
    const float* out_w     = (const float*)d_in[13];
    const float* out_b     = (const float*)d_in[14];

    float* ws = (float*)d_ws;
    float* pool1 = ws + WS_POOL1;
    float* xf    = ws + WS_XF;
    float* asum  = ws + WS_ASUM;
    float* vlada = ws + WS_VLADA;
    float* vnorm = ws + WS_VNORM;
    float* feat  = ws + WS_FEAT;

    k_zero<<<dim3((16*64 + 16*64*16 + 255) / 256), dim3(256), 0, stream>>>(ws);
    k_conv1<<<dim3(4, 64, 16), dim3(256), 0, stream>>>(img, conv1_w, conv1_b, pool1);
    k_conv2<<<dim3(2, 32, 16), dim3(256), 0, stream>>>(pool1, conv2_w, conv2_b, xf);
    k_netvlad<<<dim3(128, 16), dim3(256), 0, stream>>>(xf, nv_w, nv_b, asum, vlada);
    k_vladnorm<<<dim3(16), dim3(256), 0, stream>>>(vlada, asum, centroids, vnorm);
    k_fc<<<dim3(1), dim3(256), 0, stream>>>(vnorm, fc_w, fc_b, feat);
    k_rnn<<<dim3(1), dim3(512), 0, stream>>>(feat, cap, hid_w, hid_b, out_w, out_b,
                                             (float*)d_out);
}